// Bilaplacian_21225728377222
// MI455X (gfx1250) — compile-verified
//
#include <hip/hip_runtime.h>

typedef float v2f __attribute__((ext_vector_type(2)));
typedef float v8f __attribute__((ext_vector_type(8)));

#define HIDDEN 128
#define DIN    16
#define NOUT   8

__device__ __forceinline__ float tanh4_fast(float z) {
  // t = tanh(z) = 1 - 2/(exp(2z)+1), branch-free with native exp2/rcp.
  // exp2 overflow -> +inf -> rcp -> 0 -> t=1 ; underflow -> 0 -> t=-1.
  const float e  = __builtin_amdgcn_exp2f(z * 2.8853900817779268f); // exp(2z)
  const float t  = 1.0f - 2.0f * __builtin_amdgcn_rcpf(e + 1.0f);
  const float t2 = t * t;
  // tanh''''(z) = (16t - 24t^3)(1 - t^2)
  return (16.0f - 24.0f * t2) * t * (1.0f - t2);
}

__global__ __launch_bounds__(32)
void bilap_wmma_kernel(const float* __restrict__ x,
                       const float* __restrict__ W1,
                       const float* __restrict__ W2,
                       float* __restrict__ out) {
  // 16 batch rows x 128 hidden of G = tanh''''(z) * s_h^2, row-major
  __shared__ float Gs[16 * HIDDEN];

  const int lane = threadIdx.x;   // 0..31 (wave32)
  const int half = lane >> 4;     // K-half select per f32 WMMA layout
  const int lm   = lane & 15;
  const int koff = 2 * half;
  const int b0   = blockIdx.x * 16;
  const int row  = b0 + lm;

  // ---- A operand for GEMM1: x tile (M=16, K=16 -> 4 steps of K=4) ----
  // A-layout (32b, 16x4): lanes 0-15 M=lm K={0,1}; lanes 16-31 M=lm K={2,3}
  v2f ax[4];
#pragma unroll
  for (int kk = 0; kk < 4; ++kk) {
    const int d = 4 * kk + koff;                 // even -> 8B aligned pair
    ax[kk] = *(const v2f*)(x + row * DIN + d);
  }

  // ---- GEMM1 over 8 hidden tiles, fused tanh'''' * s^2, scatter to LDS ----
#pragma unroll
  for (int t = 0; t < 8; ++t) {
    const int h = 16 * t + lm;   // lane's hidden index: its B rows AND its C column
    v2f bw[4];
    float ssq = 0.f;
#pragma unroll
    for (int kk = 0; kk < 4; ++kk) {
      const int d = 4 * kk + koff;
      v2f w = *(const v2f*)(W1 + h * DIN + d);   // B[K=d, N=h'] = W1[h, d]
      bw[kk] = w;
      ssq += w.x * w.x + w.y * w.y;
    }
    // complete s_h: complementary K-half lives in lane^16
    ssq += __shfl_xor(ssq, 16, 32);
    const float s2 = ssq * ssq;

    v8f c = {};
#pragma unroll
    for (int kk = 0; kk < 4; ++kk) {
      c = __builtin_amdgcn_wmma_f32_16x16x4_f32(false, ax[kk], false, bw[kk],
                                                (short)0, c, false, false);
    }
    // C/D layout: VGPR r, lane -> (M = r + 8*half, N = lm); all 8 share column h
#pragma unroll
    for (int r = 0; r < 8; ++r) {
      const int m = r + 8 * half;
      Gs[m * HIDDEN + h] = tanh4_fast(c[r]) * s2;
    }
  }

  __syncthreads();   // single wave: ordering fence for LDS RAW

  // ---- GEMM2: out = G @ W2^T  (M=16, N=16 padded from 8, K=128) ----
  v8f acc = {};
#pragma unroll
  for (int kk = 0; kk < 32; ++kk) {
    const int k = 4 * kk + koff;
    v2f a2 = *(const v2f*)(&Gs[lm * HIDDEN + k]);        // A: M=lm, K={k,k+1}
    v2f b2;
    if (lm < NOUT) {                                     // B: N=lm, K={k,k+1}
      b2 = *(const v2f*)(W2 + lm * HIDDEN + k);
    } else {
      b2.x = 0.f; b2.y = 0.f;                            // pad columns 8..15
    }
    acc = __builtin_amdgcn_wmma_f32_16x16x4_f32(false, a2, false, b2,
                                                (short)0, acc, false, false);
  }

#pragma unroll
  for (int r = 0; r < 8; ++r) {
    const int m = r + 8 * half;
    if (lm < NOUT) out[(b0 + m) * NOUT + lm] = acc[r];
  }
}

extern "C" void kernel_launch(void* const* d_in, const int* in_sizes, int n_in,
                              void* d_out, int out_size, void* d_ws, size_t ws_size,
                              hipStream_t stream) {
  const float* x  = (const float*)d_in[0];   // (256, 16)
  const float* W1 = (const float*)d_in[1];   // (128, 16)
  const float* W2 = (const float*)d_in[2];   // (8, 128)
  float* out      = (float*)d_out;           // (256, 8)
  bilap_wmma_kernel<<<16, 32, 0, stream>>>(x, W1, W2, out);
}